// DSRNN_67104569032985
// MI455X (gfx1250) — compile-verified
//
#include <hip/hip_runtime.h>
#include <cstdint>
#include <cstddef>

// SNN scan for MI455X (gfx1250, wave32). fp32 WMMA (V_WMMA_F32_16X16X4_F32)
// keeps exact fp32 semantics vs the JAX reference.

typedef __attribute__((ext_vector_type(2))) float v2f;
typedef __attribute__((ext_vector_type(8))) float v8f;

#define ALPHA_ 0.9f
#define BETA_ 0.85f
#define THRESH_ 0.1f

#define B_ 64
#define T_ 256
#define D_ 512
#define N_ 1024

// ---------------------------------------------------------------------------
// Phase 0: zero the recurrent state (mem, syn, spk_in for t=0) each launch.
// ---------------------------------------------------------------------------
__global__ __launch_bounds__(256) void snn_zero(float* __restrict__ p, int n) {
  int i = blockIdx.x * blockDim.x + threadIdx.x;
  int stride = gridDim.x * blockDim.x;
  for (; i < n; i += stride) p[i] = 0.0f;
}

// ---------------------------------------------------------------------------
// Phase 1: H[b*T+t][n] = sum_d X[b][t][d] * W[d][n]   (fully parallel over T)
// X viewed as [B*T, D] row-major. One wave computes a 32x16 output block
// (2 M-tiles, shared B fragment -> half the W traffic, two independent
// WMMA chains). 8 waves per block along N.
// ---------------------------------------------------------------------------
__global__ __launch_bounds__(256) void snn_input_gemm(
    const float* __restrict__ X,   // [B*T, D]
    const float* __restrict__ W,   // [D, N]
    float* __restrict__ H)         // [B*T, N]
{
  const int lane = threadIdx.x & 31;
  const int wave = threadIdx.x >> 5;
  const int lidx = lane & 15;     // A-row / B-col within tile
  const int half = lane >> 4;     // K sub-pair select (0 -> K0,K1 ; 1 -> K2,K3)

  const int tileM = blockIdx.y * 32;
  const int tileN = (blockIdx.x * 8 + wave) * 16;

  const float* Arow0 = X + (size_t)(tileM + lidx) * D_;
  const float* Arow1 = Arow0 + (size_t)16 * D_;
  const float* Bcol  = W + tileN + lidx;

  v8f acc0 = {};
  v8f acc1 = {};
#pragma unroll 4
  for (int k = 0; k < D_; k += 4) {
    const int k0 = k + 2 * half;
    v2f b; b.x = Bcol[(size_t)k0 * N_];  b.y = Bcol[(size_t)(k0 + 1) * N_];
    v2f a0; a0.x = Arow0[k0];            a0.y = Arow0[k0 + 1];
    v2f a1; a1.x = Arow1[k0];            a1.y = Arow1[k0 + 1];
    acc0 = __builtin_amdgcn_wmma_f32_16x16x4_f32(
        false, a0, false, b, (short)0, acc0, false, false);
    acc1 = __builtin_amdgcn_wmma_f32_16x16x4_f32(
        false, a1, false, b, (short)0, acc1, false, false);
  }

#pragma unroll
  for (int r = 0; r < 8; ++r) {
    const int gRow = tileM + r + 8 * half;
    const int gCol = tileN + lidx;
    H[(size_t)gRow * N_ + gCol] = acc0[r];
    H[(size_t)(gRow + 16) * N_ + gCol] = acc1[r];
  }
}

// ---------------------------------------------------------------------------
// Phase 2 (per timestep, sequential): rec = spk_in @ R, then LIF update.
// One 16x16 output tile per block; K=1024 split across 8 waves (32 WMMA each,
// as two independent 16-deep dual-accumulator chains), reduced through LDS to
// shorten the sequential critical path of the scan.
// ---------------------------------------------------------------------------
__global__ __launch_bounds__(256) void snn_step(
    const float* __restrict__ R,        // [N, N]
    const float* __restrict__ H,        // [B*T, N]
    const float* __restrict__ spk_in,   // [B, N]
    float* __restrict__ spk_out,        // [B, N]
    float* __restrict__ mem_s,          // [B, N]
    float* __restrict__ syn_s,          // [B, N]
    float* __restrict__ out,            // concatenated outputs
    int t)
{
  __shared__ float lds[8][32 * 9];      // padded to dodge bank conflicts

  const int lane = threadIdx.x & 31;
  const int wave = threadIdx.x >> 5;    // K-slice owner: wave w -> [128w, 128w+128)
  const int lidx = lane & 15;
  const int half = lane >> 4;

  const int tileM = blockIdx.y * 16;    // batch tile
  const int tileN = blockIdx.x * 16;    // neuron tile

  const float* Arow = spk_in + (size_t)(tileM + lidx) * N_;
  const float* Bcol = R + tileN + lidx;

  const int kBeg = wave * (N_ / 8);
  const int kEnd = kBeg + (N_ / 8);

  v8f acc0 = {};
  v8f acc1 = {};
#pragma unroll 4
  for (int k = kBeg; k < kEnd; k += 8) {
    const int ka = k + 2 * half;        // even chunk
    const int kb = ka + 4;              // odd chunk (independent chain)
    v2f a0; a0.x = Arow[ka];              a0.y = Arow[ka + 1];
    v2f b0; b0.x = Bcol[(size_t)ka * N_]; b0.y = Bcol[(size_t)(ka + 1) * N_];
    v2f a1; a1.x = Arow[kb];              a1.y = Arow[kb + 1];
    v2f b1; b1.x = Bcol[(size_t)kb * N_]; b1.y = Bcol[(size_t)(kb + 1) * N_];
    acc0 = __builtin_amdgcn_wmma_f32_16x16x4_f32(
        false, a0, false, b0, (short)0, acc0, false, false);
    acc1 = __builtin_amdgcn_wmma_f32_16x16x4_f32(
        false, a1, false, b1, (short)0, acc1, false, false);
  }

#pragma unroll
  for (int r = 0; r < 8; ++r) lds[wave][lane * 9 + r] = acc0[r] + acc1[r];
  __syncthreads();

  // Output regions (reference return order, flattened):
  // spk_rec[B,T,N] | mem[B,N] | syn[B,N] | spk[B,N] | mem_rec[B,T,N] | syn_rec[B,T,N]
  const size_t recN = (size_t)B_ * T_ * N_;
  const size_t stN  = (size_t)B_ * N_;
  float* out_spk_rec = out;
  float* out_mem     = out + recN;
  float* out_syn     = out + recN + stN;
  float* out_spk     = out + recN + 2 * stN;
  float* out_mem_rec = out + recN + 3 * stN;
  float* out_syn_rec = out + 2 * recN + 3 * stN;

  // 256 tile elements, 256 threads -> 1 each.
  {
    const int e = threadIdx.x;
    const int elane = e & 31;
    const int er    = e >> 5;
    float rec = 0.0f;
#pragma unroll
    for (int w = 0; w < 8; ++w) rec += lds[w][elane * 9 + er];

    const int row = er + 8 * (elane >> 4);
    const int col = elane & 15;
    const int b = tileM + row;
    const int n = tileN + col;
    const size_t sidx = (size_t)b * N_ + n;                 // state index
    const size_t ridx = ((size_t)b * T_ + t) * N_ + n;      // record index

    const float h       = H[ridx];
    const float syn_new = ALPHA_ * syn_s[sidx] + h + rec;
    const float mem_pre = BETA_  * mem_s[sidx] + syn_new;
    const float mthr    = mem_pre - THRESH_;
    const float spike   = (mthr > 0.0f) ? 1.0f : 0.0f;
    const float mem_new = (mthr > 0.0f) ? 0.0f : mem_pre;

    out_spk_rec[ridx] = spike;
    out_mem_rec[ridx] = mem_new;
    out_syn_rec[ridx] = syn_new;

    syn_s[sidx]   = syn_new;
    mem_s[sidx]   = mem_new;
    spk_out[sidx] = spike;

    if (t == T_ - 1) {            // final carry
      out_mem[sidx] = mem_new;
      out_syn[sidx] = syn_new;
      out_spk[sidx] = spike;
    }
  }
}

// ---------------------------------------------------------------------------
extern "C" void kernel_launch(void* const* d_in, const int* in_sizes, int n_in,
                              void* d_out, int out_size, void* d_ws, size_t ws_size,
                              hipStream_t stream) {
  (void)in_sizes; (void)n_in; (void)out_size; (void)ws_size;

  const float* X = (const float*)d_in[0];   // [B, T, D]
  const float* W = (const float*)d_in[1];   // [D, N]
  const float* R = (const float*)d_in[2];   // [N, N]
  float* out = (float*)d_out;

  // Workspace layout (floats): H[T*B*N] | spk_a | mem_s | syn_s | spk_b
  float* H     = (float*)d_ws;
  float* spk_a = H + (size_t)B_ * T_ * N_;
  float* mem_s = spk_a + (size_t)B_ * N_;
  float* syn_s = mem_s + (size_t)B_ * N_;
  float* spk_b = syn_s + (size_t)B_ * N_;

  // Zero spk_a + mem_s + syn_s (contiguous) every launch for replay determinism.
  const int zeroN = 3 * B_ * N_;
  snn_zero<<<dim3((zeroN + 255) / 256), dim3(256), 0, stream>>>(spk_a, zeroN);

  // Phase 1: H = X @ W over all timesteps (parallel).
  snn_input_gemm<<<dim3(N_ / 128, (B_ * T_) / 32), dim3(256), 0, stream>>>(X, W, H);

  // Phase 2: sequential scan; one launch per timestep, spike double-buffered.
  for (int t = 0; t < T_; ++t) {
    const float* spk_in = (t & 1) ? spk_b : spk_a;
    float* spk_out      = (t & 1) ? spk_a : spk_b;
    snn_step<<<dim3(N_ / 16, B_ / 16), dim3(256), 0, stream>>>(
        R, H, spk_in, spk_out, mem_s, syn_s, out, t);
  }
}